// Net_16561393893563
// MI455X (gfx1250) — compile-verified
//
#include <hip/hip_runtime.h>

// ---------------------------------------------------------------------------
// 2-layer GCN (PyG GCNConv semantics) + mean pooling, for MI455X / gfx1250.
// Dense transforms use V_WMMA_F32_16X16X4_F32 (full fp32 precision).
// Edge aggregation is L2-resident atomic scatter-add (workset ~32MB << 192MB L2);
// gathers are forced to GLOBAL_LOAD_B128 via float4 (rows are 16B-aligned).
// ---------------------------------------------------------------------------

typedef __attribute__((ext_vector_type(2))) float v2f;
typedef __attribute__((ext_vector_type(8))) float v8f;

#define WMMA_F32_4(A, B, C) \
  __builtin_amdgcn_wmma_f32_16x16x4_f32(false, (A), false, (B), (short)0, (C), false, false)

// ----------------------------- utility kernels -----------------------------

__global__ void fill_f32(float* __restrict__ p, float v, int n) {
  int i = blockIdx.x * blockDim.x + threadIdx.x;
  if (i < n) p[i] = v;
}

__global__ void degree_count(const long long* __restrict__ dst,
                             float* __restrict__ deg, int E) {
  int e = blockIdx.x * blockDim.x + threadIdx.x;
  if (e < E) atomicAdd(&deg[(int)dst[e]], 1.0f);
}

__global__ void make_dinv(float* __restrict__ deg, int N) {
  int n = blockIdx.x * blockDim.x + threadIdx.x;
  if (n < N) {
    float d = deg[n];
    deg[n] = (d > 0.0f) ? (1.0f / sqrtf(d)) : 0.0f;
  }
}

// ------------------------ layer 1: hs1 = (x @ W1) * dinv[row] --------------
// x: [N,5]  W1: [5,30]  out hs1: [N,32] (cols 30,31 are exact zeros)
// One wave (32 threads) per 16-row tile. K padded 5->8 (two WMMA K-steps),
// N padded 30->32 (two 16-col tiles).

__global__ void gemm1_wmma(const float* __restrict__ x, const float* __restrict__ W1,
                           const float* __restrict__ dinv, float* __restrict__ hs1,
                           int N) {
  const int lane = threadIdx.x;      // 0..31
  const int half = lane >> 4;        // 0: K pair {0,1}/{4,5}; 1: {2,3}/{6,7}
  const int l    = lane & 15;
  const int m0   = blockIdx.x * 16;
  const int arow = m0 + l;
  const bool rv  = arow < N;
  const int kb   = half * 2;

  // A tile (16x4 per K-step), fp32, padded with zeros past K=4
  v2f a0, a1;
  a0.x = rv ? x[arow * 5 + kb]     : 0.0f;                    // k = 0 / 2
  a0.y = rv ? x[arow * 5 + kb + 1] : 0.0f;                    // k = 1 / 3
  a1.x = (rv && half == 0) ? x[arow * 5 + 4] : 0.0f;          // k = 4 (k=6 -> 0)
  a1.y = 0.0f;                                                // k = 5 / 7

  // B tiles (4x16 per K-step) for N-tile 0 (cols 0..15) and 1 (cols 16..31)
  int n0 = l;
  int n1 = l + 16;
  bool nv1 = n1 < 30;
  v2f b00, b01, b10, b11;
  b00.x = W1[kb * 30 + n0];
  b00.y = W1[(kb + 1) * 30 + n0];
  b01.x = (half == 0) ? W1[4 * 30 + n0] : 0.0f;
  b01.y = 0.0f;
  b10.x = nv1 ? W1[kb * 30 + n1]       : 0.0f;
  b10.y = nv1 ? W1[(kb + 1) * 30 + n1] : 0.0f;
  b11.x = (nv1 && half == 0) ? W1[4 * 30 + n1] : 0.0f;
  b11.y = 0.0f;

  v8f c0 = {0, 0, 0, 0, 0, 0, 0, 0};
  v8f c1 = {0, 0, 0, 0, 0, 0, 0, 0};
  c0 = WMMA_F32_4(a0, b00, c0);
  c0 = WMMA_F32_4(a1, b01, c0);
  c1 = WMMA_F32_4(a0, b10, c1);
  c1 = WMMA_F32_4(a1, b11, c1);

  // Epilogue: scale each row by dinv[row] and store (cols 30,31 compute to 0).
  // Fast path: full tile (always taken when N % 16 == 0) -> no per-row branches.
  if (m0 + 16 <= N) {
#pragma unroll
    for (int v = 0; v < 8; ++v) {
      int m = m0 + v + half * 8;
      float dm = dinv[m];
      hs1[(size_t)m * 32 + l]      = c0[v] * dm;
      hs1[(size_t)m * 32 + 16 + l] = c1[v] * dm;
    }
  } else {
#pragma unroll
    for (int v = 0; v < 8; ++v) {
      int m = m0 + v + half * 8;
      if (m < N) {
        float dm = dinv[m];
        hs1[(size_t)m * 32 + l]      = c0[v] * dm;
        hs1[(size_t)m * 32 + 16 + l] = c1[v] * dm;
      }
    }
  }
}

// ------------------------ layer 2: hs2 = (y1 @ W2) * dinv[row] -------------
// y1: [N,32] (cols 30,31 zero)  W2: [30,8]  out hs2: [N,8]
// K padded 30->32 (8 WMMA K-steps), single N tile with cols 8..15 zero.

__global__ void gemm2_wmma(const float* __restrict__ y1, const float* __restrict__ W2,
                           const float* __restrict__ dinv, float* __restrict__ hs2,
                           int N) {
  const int lane = threadIdx.x;
  const int half = lane >> 4;
  const int l    = lane & 15;
  const int m0   = blockIdx.x * 16;
  const int arow = m0 + l;
  const bool rv  = arow < N;

  v8f c = {0, 0, 0, 0, 0, 0, 0, 0};
#pragma unroll
  for (int s = 0; s < 8; ++s) {
    int k0 = 4 * s + half * 2;
    v2f a, b;
    a.x = rv ? y1[(size_t)arow * 32 + k0]     : 0.0f;
    a.y = rv ? y1[(size_t)arow * 32 + k0 + 1] : 0.0f;
    bool nv = l < 8;
    b.x = (nv && (k0     < 30)) ? W2[k0 * 8 + l]       : 0.0f;
    b.y = (nv && (k0 + 1 < 30)) ? W2[(k0 + 1) * 8 + l] : 0.0f;
    c = WMMA_F32_4(a, b, c);
  }

  if (m0 + 16 <= N) {
    if (l < 8) {
#pragma unroll
      for (int v = 0; v < 8; ++v) {
        int m = m0 + v + half * 8;
        hs2[(size_t)m * 8 + l] = c[v] * dinv[m];
      }
    }
  } else {
#pragma unroll
    for (int v = 0; v < 8; ++v) {
      int m = m0 + v + half * 8;
      if (m < N && l < 8) {
        hs2[(size_t)m * 8 + l] = c[v] * dinv[m];
      }
    }
  }
}

// --------------------------- edge scatter-add ------------------------------
// hs rows are 128B-strided and 16B-aligned -> force GLOBAL_LOAD_B128 gathers.

__global__ void scatter_edges30(const long long* __restrict__ src,
                                const long long* __restrict__ dst,
                                const float* __restrict__ hs,
                                float* __restrict__ agg, int E) {
  int e = blockIdx.x * blockDim.x + threadIdx.x;
  if (e >= E) return;
  int s = (int)src[e];
  int d = (int)dst[e];
  const float4* hp4 = reinterpret_cast<const float4*>(hs + (size_t)s * 32);
  float* ap = agg + (size_t)d * 32;

  float4 r[8];
#pragma unroll
  for (int q = 0; q < 8; ++q) r[q] = hp4[q];   // 8x b128 = cols 0..31 (30,31 zero)

#pragma unroll
  for (int q = 0; q < 7; ++q) {
    atomicAdd(&ap[q * 4 + 0], r[q].x);
    atomicAdd(&ap[q * 4 + 1], r[q].y);
    atomicAdd(&ap[q * 4 + 2], r[q].z);
    atomicAdd(&ap[q * 4 + 3], r[q].w);
  }
  atomicAdd(&ap[28], r[7].x);
  atomicAdd(&ap[29], r[7].y);
}

__global__ void scatter_edges8(const long long* __restrict__ src,
                               const long long* __restrict__ dst,
                               const float* __restrict__ hs,
                               float* __restrict__ agg, int E) {
  int e = blockIdx.x * blockDim.x + threadIdx.x;
  if (e >= E) return;
  int s = (int)src[e];
  int d = (int)dst[e];
  const float4* hp4 = reinterpret_cast<const float4*>(hs + (size_t)s * 8);
  float* ap = agg + (size_t)d * 8;

  float4 r0 = hp4[0];
  float4 r1 = hp4[1];
  atomicAdd(&ap[0], r0.x);
  atomicAdd(&ap[1], r0.y);
  atomicAdd(&ap[2], r0.z);
  atomicAdd(&ap[3], r0.w);
  atomicAdd(&ap[4], r1.x);
  atomicAdd(&ap[5], r1.y);
  atomicAdd(&ap[6], r1.z);
  atomicAdd(&ap[7], r1.w);
}

// --------------------------- finalize kernels ------------------------------
// y1 = relu(dinv * (agg1 + hs1) + b1), in place over agg1 (self-loop folded in)

__global__ void finalize1(float* __restrict__ agg1, const float* __restrict__ hs1,
                          const float* __restrict__ dinv, const float* __restrict__ b1,
                          int N) {
  int idx = blockIdx.x * blockDim.x + threadIdx.x;
  if (idx >= N * 32) return;
  int n = idx >> 5;
  int c = idx & 31;
  if (c < 30) {
    float v = dinv[n] * (agg1[idx] + hs1[idx]) + b1[c];
    agg1[idx] = fmaxf(v, 0.0f);
  }
  // cols 30,31 stay 0 from the fill pass (needed as K-padding for gemm2)
}

// h2 = dinv * (agg2 + hs2) + b2, then scatter-mean accumulation per graph

__global__ void finalize2_pool(const float* __restrict__ agg2, const float* __restrict__ hs2,
                               const float* __restrict__ dinv, const float* __restrict__ b2,
                               const long long* __restrict__ batch,
                               float* __restrict__ psum, float* __restrict__ pcnt, int N) {
  int n = blockIdx.x * blockDim.x + threadIdx.x;
  if (n >= N) return;
  int g = (int)batch[n];
  float di = dinv[n];
#pragma unroll
  for (int c = 0; c < 8; ++c) {
    float v = di * (agg2[(size_t)n * 8 + c] + hs2[(size_t)n * 8 + c]) + b2[c];
    atomicAdd(&psum[g * 8 + c], v);
  }
  atomicAdd(&pcnt[g], 1.0f);
}

__global__ void pool_div(const float* __restrict__ psum, const float* __restrict__ pcnt,
                         float* __restrict__ out, int total) {
  int i = blockIdx.x * blockDim.x + threadIdx.x;
  if (i >= total) return;
  out[i] = psum[i] / fmaxf(pcnt[i >> 3], 1.0f);
}

// ------------------------------- launcher ----------------------------------

extern "C" void kernel_launch(void* const* d_in, const int* in_sizes, int n_in,
                              void* d_out, int out_size, void* d_ws, size_t ws_size,
                              hipStream_t stream) {
  const float*     x     = (const float*)d_in[0];
  const long long* ei    = (const long long*)d_in[1];   // [2, E] int64
  const long long* batch = (const long long*)d_in[2];   // [N] int64
  const float*     W1    = (const float*)d_in[3];       // [5,30]
  const float*     b1    = (const float*)d_in[4];       // [30]
  const float*     W2    = (const float*)d_in[5];       // [30,8]
  const float*     b2    = (const float*)d_in[6];       // [8]

  const int N = in_sizes[0] / 5;
  const int E = in_sizes[1] / 2;
  const int G = out_size / 8;

  const long long* src = ei;
  const long long* dst = ei + E;

  // Workspace layout (floats); total ~32.5 MB -> L2 resident.
  float* ws   = (float*)d_ws;
  float* dinv = ws;                          // N     (deg, then deg^-1/2)
  float* hs1  = dinv + N;                    // N*32  (h1 * dinv[row], padded)
  float* agg1 = hs1 + (size_t)N * 32;        // N*32  (edge sums; becomes y1)
  float* hs2  = agg1 + (size_t)N * 32;       // N*8
  float* agg2 = hs2 + (size_t)N * 8;         // N*8
  float* psum = agg2 + (size_t)N * 8;        // G*8
  float* pcnt = psum + (size_t)G * 8;        // G   (contiguous with psum)

  const int B = 256;
  // Re-initialize all accumulators every call (harness does not re-poison).
  fill_f32<<<(N + B - 1) / B, B, 0, stream>>>(dinv, 1.0f, N);  // deg starts at 1 (self-loop)
  fill_f32<<<(N * 32 + B - 1) / B, B, 0, stream>>>(agg1, 0.0f, N * 32);
  fill_f32<<<(N * 8 + B - 1) / B, B, 0, stream>>>(agg2, 0.0f, N * 8);
  fill_f32<<<(G * 9 + B - 1) / B, B, 0, stream>>>(psum, 0.0f, G * 9);  // psum+pcnt

  // Degrees -> dinv
  degree_count<<<(E + B - 1) / B, B, 0, stream>>>(dst, dinv, E);
  make_dinv<<<(N + B - 1) / B, B, 0, stream>>>(dinv, N);

  const int mTiles = (N + 15) / 16;  // one wave per 16-row tile

  // Layer 1
  gemm1_wmma<<<mTiles, 32, 0, stream>>>(x, W1, dinv, hs1, N);
  scatter_edges30<<<(E + B - 1) / B, B, 0, stream>>>(src, dst, hs1, agg1, E);
  finalize1<<<(N * 32 + B - 1) / B, B, 0, stream>>>(agg1, hs1, dinv, b1, N);

  // Layer 2 (reads y1 == agg1 in place)
  gemm2_wmma<<<mTiles, 32, 0, stream>>>(agg1, W2, dinv, hs2, N);
  scatter_edges8<<<(E + B - 1) / B, B, 0, stream>>>(src, dst, hs2, agg2, E);

  // Finalize + mean pooling
  finalize2_pool<<<(N + B - 1) / B, B, 0, stream>>>(agg2, hs2, dinv, b2, batch,
                                                    psum, pcnt, N);
  pool_div<<<(G * 8 + B - 1) / B, B, 0, stream>>>(psum, pcnt, (float*)d_out, G * 8);
}